// ConvGuidedFilter_58634893525618
// MI455X (gfx1250) — compile-verified
//
#include <hip/hip_runtime.h>

typedef __attribute__((ext_vector_type(2))) float v2f;
typedef __attribute__((ext_vector_type(4))) float v4f;
typedef __attribute__((ext_vector_type(8))) float v8f;

#define P_TOT   262144      // 16 * 128 * 128
#define LRH     128
#define LRW     128
#define HRH     1024
#define HRW     1024
#define NBATCH  16
#define GBLK    2048        // P_TOT / 128 rows per gemm block

__device__ __forceinline__ v8f wmma_k4(v2f a, v2f b, v8f c) {
  // V_WMMA_F32_16X16X4_F32 : D = A(16x4) * B(4x16) + C
  return __builtin_amdgcn_wmma_f32_16x16x4_f32(false, a, false, b, (short)0, c,
                                               false, false);
}

// ---------------------------------------------------------------------------
// K1: guided-filter features on low-res.
//   feat[p*6 + c]   = cov_xy,  feat[p*6 + 3 + c] = var_x
//   meanx[c*P + p], meany[c*P + p]   (planar for later use)
// ---------------------------------------------------------------------------
__global__ void __launch_bounds__(256)
k_feat(const float* __restrict__ img, const float* __restrict__ gui,
       float* __restrict__ feat, float* __restrict__ meanx,
       float* __restrict__ meany) {
  int p = blockIdx.x * 256 + threadIdx.x;
  if (p >= P_TOT) return;
  int x = p & (LRW - 1);
  int y = (p >> 7) & (LRH - 1);
  int b = p >> 14;
  int x0 = max(x - 1, 0), x1 = min(x + 1, LRW - 1);
  int y0 = max(y - 1, 0), y1 = min(y + 1, LRH - 1);
  float n = (float)((x1 - x0 + 1) * (y1 - y0 + 1));
#pragma unroll
  for (int c = 0; c < 3; ++c) {
    const float* ib = img + (size_t)((b * 3 + c) * LRH) * LRW;
    const float* gb = gui + (size_t)((b * 3 + c) * LRH) * LRW;
    float si = 0.f, sg = 0.f, sig = 0.f, sii = 0.f;
    for (int yy = y0; yy <= y1; ++yy)
      for (int xx = x0; xx <= x1; ++xx) {
        float iv = ib[yy * LRW + xx];
        float gv = gb[yy * LRW + xx];
        si += iv; sg += gv; sig += iv * gv; sii += iv * iv;
      }
    float mx = si / n;
    float my = sg / n;
    feat[(size_t)p * 6 + c]     = sig / n - mx * my;   // cov_xy
    feat[(size_t)p * 6 + 3 + c] = sii / n - mx * mx;   // var_x
    meanx[(size_t)c * P_TOT + p] = mx;
    meany[(size_t)c * P_TOT + p] = my;
  }
}

// ---------------------------------------------------------------------------
// K2: y1 = feat[P,6] @ w1^T[6,32] via WMMA f32 16x16x4 (K padded to 8).
// Padded K lanes load unconditionally (always in-bounds) and select zero,
// avoiding exec-mask-predicated loads.
// ---------------------------------------------------------------------------
__global__ void __launch_bounds__(256)
k_gemm1(const float* __restrict__ feat, const float* __restrict__ w1,
        float* __restrict__ y1, float* __restrict__ part) {
  __shared__ float sm_s[8][32];
  __shared__ float sm_q[8][32];
  const int lane = threadIdx.x & 31;
  const int wv   = threadIdx.x >> 5;
  const int r    = lane & 15;
  const int hi   = lane >> 4;
  const int row0 = blockIdx.x * 128 + wv * 16;

  const float* arow = feat + (size_t)(row0 + r) * 6;
  float a4 = arow[4], a5 = arow[5];            // unconditional loads
  v2f a0, a1;
  a0.x = arow[2 * hi];
  a0.y = arow[2 * hi + 1];
  a1.x = hi ? 0.f : a4;                        // v_cndmask, no branch
  a1.y = hi ? 0.f : a5;

  float s_t[2], q_t[2];
#pragma unroll
  for (int t = 0; t < 2; ++t) {
    int c = t * 16 + r;
    float w4 = w1[c * 6 + 4], w5 = w1[c * 6 + 5];
    v2f b0, b1;
    b0.x = w1[c * 6 + 2 * hi];
    b0.y = w1[c * 6 + 2 * hi + 1];
    b1.x = hi ? 0.f : w4;
    b1.y = hi ? 0.f : w5;
    v8f acc = {0.f, 0.f, 0.f, 0.f, 0.f, 0.f, 0.f, 0.f};
    acc = wmma_k4(a0, b0, acc);
    acc = wmma_k4(a1, b1, acc);
    float s = 0.f, q = 0.f;
#pragma unroll
    for (int i = 0; i < 8; ++i) {
      float v = acc[i];
      y1[(size_t)(row0 + i + 8 * hi) * 32 + c] = v;
      s += v; q += v * v;
    }
    s += __shfl_down(s, 16, 32);
    q += __shfl_down(q, 16, 32);
    s_t[t] = s; q_t[t] = q;
  }
  if (hi == 0) {
    sm_s[wv][r] = s_t[0]; sm_s[wv][16 + r] = s_t[1];
    sm_q[wv][r] = q_t[0]; sm_q[wv][16 + r] = q_t[1];
  }
  __syncthreads();
  if (threadIdx.x < 64) {
    int c = threadIdx.x & 31;
    bool isq = threadIdx.x >= 32;
    float acc = 0.f;
#pragma unroll
    for (int w = 0; w < 8; ++w) acc += isq ? sm_q[w][c] : sm_s[w][c];
    part[(size_t)blockIdx.x * 64 + (isq ? 32 : 0) + c] = acc;
  }
}

// ---------------------------------------------------------------------------
// K3/K5: deterministic reduction of per-block partials -> BN scale/shift.
//   ss[c] = g*rsqrt(var+eps), ss[32+c] = beta - mean*ss[c]
// ---------------------------------------------------------------------------
__global__ void __launch_bounds__(256)
k_bnstats(const float* __restrict__ part, int nblocks,
          const float* __restrict__ gamma, const float* __restrict__ beta,
          float invP, float* __restrict__ ss) {
  __shared__ float sm[4][64];
  int slot = threadIdx.x & 63;       // 0..31 sums, 32..63 sumsq
  int g    = threadIdx.x >> 6;       // 4 groups
  float a0 = 0.f, a1 = 0.f, a2 = 0.f, a3 = 0.f;
  for (int blk = g * 4; blk < nblocks; blk += 16) {
    a0 += part[(size_t)(blk + 0) * 64 + slot];
    a1 += part[(size_t)(blk + 1) * 64 + slot];
    a2 += part[(size_t)(blk + 2) * 64 + slot];
    a3 += part[(size_t)(blk + 3) * 64 + slot];
  }
  sm[g][slot] = ((a0 + a1) + (a2 + a3));
  __syncthreads();
  if (threadIdx.x < 32) {
    int c = threadIdx.x;
    float S = 0.f, Q = 0.f;
#pragma unroll
    for (int gg = 0; gg < 4; ++gg) { S += sm[gg][c]; Q += sm[gg][32 + c]; }
    float mean = S * invP;
    float var  = Q * invP - mean * mean;
    float sc   = gamma[c] * rsqrtf(var + 1e-5f);
    ss[c]      = sc;
    ss[32 + c] = beta[c] - mean * sc;
  }
}

// ---------------------------------------------------------------------------
// K4: y2 = relu(bn(y1)) @ w2^T  (K=32 -> 8 WMMA k-steps), + BN partials.
// ---------------------------------------------------------------------------
__global__ void __launch_bounds__(256)
k_gemm2(const float* __restrict__ y1, const float* __restrict__ ss1,
        const float* __restrict__ w2, float* __restrict__ y2,
        float* __restrict__ part) {
  __shared__ float sm_s[8][32];
  __shared__ float sm_q[8][32];
  const int lane = threadIdx.x & 31;
  const int wv   = threadIdx.x >> 5;
  const int r    = lane & 15;
  const int hi   = lane >> 4;
  const int row0 = blockIdx.x * 128 + wv * 16;

  const float* arow = y1 + (size_t)(row0 + r) * 32;
  v2f a[8];
#pragma unroll
  for (int ks = 0; ks < 8; ++ks) {
    int k = ks * 4 + 2 * hi;
    a[ks].x = fmaxf(arow[k]     * ss1[k]     + ss1[32 + k],     0.f);
    a[ks].y = fmaxf(arow[k + 1] * ss1[k + 1] + ss1[32 + k + 1], 0.f);
  }
  float s_t[2], q_t[2];
#pragma unroll
  for (int t = 0; t < 2; ++t) {
    int c = t * 16 + r;
    v2f b[8];
#pragma unroll
    for (int ks = 0; ks < 8; ++ks) {
      int k = ks * 4 + 2 * hi;
      b[ks].x = w2[c * 32 + k];
      b[ks].y = w2[c * 32 + k + 1];
    }
    v8f acc = {0.f, 0.f, 0.f, 0.f, 0.f, 0.f, 0.f, 0.f};
#pragma unroll
    for (int ks = 0; ks < 8; ++ks) acc = wmma_k4(a[ks], b[ks], acc);
    float s = 0.f, q = 0.f;
#pragma unroll
    for (int i = 0; i < 8; ++i) {
      float v = acc[i];
      y2[(size_t)(row0 + i + 8 * hi) * 32 + c] = v;
      s += v; q += v * v;
    }
    s += __shfl_down(s, 16, 32);
    q += __shfl_down(q, 16, 32);
    s_t[t] = s; q_t[t] = q;
  }
  if (hi == 0) {
    sm_s[wv][r] = s_t[0]; sm_s[wv][16 + r] = s_t[1];
    sm_q[wv][r] = q_t[0]; sm_q[wv][16 + r] = q_t[1];
  }
  __syncthreads();
  if (threadIdx.x < 64) {
    int c = threadIdx.x & 31;
    bool isq = threadIdx.x >= 32;
    float acc = 0.f;
#pragma unroll
    for (int w = 0; w < 8; ++w) acc += isq ? sm_q[w][c] : sm_s[w][c];
    part[(size_t)blockIdx.x * 64 + (isq ? 32 : 0) + c] = acc;
  }
}

// ---------------------------------------------------------------------------
// K6: A = relu(bn(y2)) @ w3^T (N=3 padded to 16); b = mean_y - A*mean_x.
// w3 loads use clamped index + select (no predicated loads).
// ---------------------------------------------------------------------------
__global__ void __launch_bounds__(256)
k_gemm3(const float* __restrict__ y2, const float* __restrict__ ss2,
        const float* __restrict__ w3, const float* __restrict__ meanx,
        const float* __restrict__ meany, float* __restrict__ Abuf,
        float* __restrict__ bbuf) {
  const int lane = threadIdx.x & 31;
  const int wv   = threadIdx.x >> 5;
  const int r    = lane & 15;
  const int hi   = lane >> 4;
  const int row0 = blockIdx.x * 128 + wv * 16;
  const int rc   = (r < 3) ? r : 2;            // clamped, always valid

  const float* arow = y2 + (size_t)(row0 + r) * 32;
  v2f a[8], b[8];
#pragma unroll
  for (int ks = 0; ks < 8; ++ks) {
    int k = ks * 4 + 2 * hi;
    a[ks].x = fmaxf(arow[k]     * ss2[k]     + ss2[32 + k],     0.f);
    a[ks].y = fmaxf(arow[k + 1] * ss2[k + 1] + ss2[32 + k + 1], 0.f);
    float wx = w3[rc * 32 + k];                // unconditional loads
    float wy = w3[rc * 32 + k + 1];
    b[ks].x = (r < 3) ? wx : 0.f;              // v_cndmask
    b[ks].y = (r < 3) ? wy : 0.f;
  }
  v8f acc = {0.f, 0.f, 0.f, 0.f, 0.f, 0.f, 0.f, 0.f};
#pragma unroll
  for (int ks = 0; ks < 8; ++ks) acc = wmma_k4(a[ks], b[ks], acc);

  if (r < 3) {
#pragma unroll
    for (int i = 0; i < 8; ++i) {
      int row = row0 + i + 8 * hi;
      size_t idx = (size_t)r * P_TOT + row;
      float Av = acc[i];
      Abuf[idx] = Av;
      bbuf[idx] = meany[idx] - Av * meanx[idx];
    }
  }
}

// ---------------------------------------------------------------------------
// K7: fused align_corners bilinear upsample of A,b + out = A*hr + b.
// Streaming HR traffic (201 MB in + 201 MB out) is non-temporal so the
// reused A/b taps (~6 MB) stay resident in WGP$/L2.
// ---------------------------------------------------------------------------
__global__ void __launch_bounds__(256)
k_apply(const float* __restrict__ Abuf, const float* __restrict__ bbuf,
        const float* __restrict__ hr, float* __restrict__ out) {
  int idx = blockIdx.x * 256 + threadIdx.x;     // (b,c,yy,x4), x4 fastest
  int x4 = idx & 255;
  int t  = idx >> 8;
  int yy = t & 1023;
  int bc = t >> 10;                              // b*3 + c
  int c  = bc % 3;
  int b  = bc / 3;

  const float SC = 127.0f / 1023.0f;             // (h-1)/(H-1), align_corners
  float cy = yy * SC;
  int   i0 = (int)cy; if (i0 > 126) i0 = 126;
  float th = cy - (float)i0;

  size_t base = (size_t)c * P_TOT + (size_t)b * (LRH * LRW) + (size_t)i0 * LRW;
  const float* A0 = Abuf + base; const float* A1 = A0 + LRW;
  const float* B0 = bbuf + base; const float* B1 = B0 + LRW;

  v4f hv = __builtin_nontemporal_load((const v4f*)hr + idx);
  v4f ov;
  int xb = x4 * 4;
#pragma unroll
  for (int j = 0; j < 4; ++j) {
    float cx = (float)(xb + j) * SC;
    int   j0 = (int)cx; if (j0 > 126) j0 = 126;
    float tw = cx - (float)j0;
    float a = (A0[j0] * (1.f - tw) + A0[j0 + 1] * tw) * (1.f - th) +
              (A1[j0] * (1.f - tw) + A1[j0 + 1] * tw) * th;
    float bb = (B0[j0] * (1.f - tw) + B0[j0 + 1] * tw) * (1.f - th) +
               (B1[j0] * (1.f - tw) + B1[j0 + 1] * tw) * th;
    ov[j] = a * hv[j] + bb;
  }
  __builtin_nontemporal_store(ov, (v4f*)out + idx);
}

// ---------------------------------------------------------------------------
extern "C" void kernel_launch(void* const* d_in, const int* in_sizes, int n_in,
                              void* d_out, int out_size, void* d_ws,
                              size_t ws_size, hipStream_t stream) {
  const float* image_lr = (const float*)d_in[0];
  const float* guide_lr = (const float*)d_in[1];
  const float* image_hr = (const float*)d_in[2];
  // d_in[3] = w_box (all-ones, folded into k_feat)
  const float* w1 = (const float*)d_in[4];
  const float* g1 = (const float*)d_in[5];
  const float* b1 = (const float*)d_in[6];
  const float* w2 = (const float*)d_in[7];
  const float* g2 = (const float*)d_in[8];
  const float* b2 = (const float*)d_in[9];
  const float* w3 = (const float*)d_in[10];
  float* out = (float*)d_out;

  float* ws    = (float*)d_ws;
  float* feat  = ws;                              // 6P (reused as A,b below)
  float* meanx = ws + (size_t)6 * P_TOT;          // 3P
  float* meany = meanx + (size_t)3 * P_TOT;       // 3P
  float* y1    = meany + (size_t)3 * P_TOT;       // 32P
  float* y2    = y1 + (size_t)32 * P_TOT;         // 32P
  float* part1 = y2 + (size_t)32 * P_TOT;         // GBLK*64
  float* part2 = part1 + (size_t)GBLK * 64;       // GBLK*64
  float* ss1   = part2 + (size_t)GBLK * 64;       // 64
  float* ss2   = ss1 + 64;                        // 64
  float* Abuf  = feat;                            // 3P (feat dead after K2)
  float* bbuf  = feat + (size_t)3 * P_TOT;        // 3P

  const float invP = 1.0f / (float)P_TOT;

  k_feat<<<P_TOT / 256, 256, 0, stream>>>(image_lr, guide_lr, feat, meanx,
                                          meany);
  k_gemm1<<<GBLK, 256, 0, stream>>>(feat, w1, y1, part1);
  k_bnstats<<<1, 256, 0, stream>>>(part1, GBLK, g1, b1, invP, ss1);
  k_gemm2<<<GBLK, 256, 0, stream>>>(y1, ss1, w2, y2, part2);
  k_bnstats<<<1, 256, 0, stream>>>(part2, GBLK, g2, b2, invP, ss2);
  k_gemm3<<<GBLK, 256, 0, stream>>>(y2, ss2, w3, meanx, meany, Abuf, bbuf);
  k_apply<<<(NBATCH * 3 * HRH * (HRW / 4)) / 256, 256, 0, stream>>>(
      Abuf, bbuf, image_hr, out);
}